// ChainCRF_87290915324245
// MI455X (gfx1250) — compile-verified
//
#include <hip/hip_runtime.h>

// Chain-CRF loss for B=128, T=2048, C=32 on gfx1250 (MI455X).
// Kernel 1: gold-path energy (parallel gathers) -> d_ws.
// Kernel 2: sequential log-semiring scan, one wave per 16 batch rows,
//           each step = exp-shift + [16x32]x[32x32] fp32 GEMM via
//           V_WMMA_F32_16X16X4_F32 + raw v_log_f32, out = log_z - energy.

#define B_  128
#define T_  2048
#define C_  32
#define TC_ (T_ * C_)

#define LN2F 0.6931471805599453f

typedef float v2f __attribute__((ext_vector_type(2)));
typedef float v8f __attribute__((ext_vector_type(8)));

#if defined(__AMDGCN__) && __has_builtin(__builtin_amdgcn_wmma_f32_16x16x4_f32)
#define HAVE_WMMA_F32X4 1
#else
#define HAVE_WMMA_F32X4 0
#endif

// Raw hardware log2 (v_log_f32). Safe here: arguments are bounded well away
// from the denormal range (w >= min(exp(U)) since max-shifted p has a 1.0).
__device__ __forceinline__ float fast_ln(float w) {
#if defined(__AMDGCN__) && __has_builtin(__builtin_amdgcn_logf)
    return __builtin_amdgcn_logf(w) * LN2F;
#else
    return __logf(w);
#endif
}

// ---------------------------------------------------------------------------
// Kernel 1: gold-path energy. One wave per batch element, lanes stride T.
// ---------------------------------------------------------------------------
__global__ __launch_bounds__(256) void crf_energy_kernel(
    const float* __restrict__ x, const float* __restrict__ U,
    const float* __restrict__ b_start, const float* __restrict__ b_end,
    const int* __restrict__ y, float* __restrict__ energy)
{
    const int wave = threadIdx.x >> 5;
    const int lane = threadIdx.x & 31;
    const int b = blockIdx.x * 8 + wave;
    if (b >= B_) return;

    const int*   yb = y + (size_t)b * T_;
    const float* xb = x + (size_t)b * TC_;

    float e = 0.f;
    for (int t = lane; t < T_; t += 32) {
        const int yt = yb[t];
        e += xb[t * C_ + yt];
        if (t < T_ - 1) e += U[yt * C_ + yb[t + 1]];
    }
    if (lane == 0) e += b_start[yb[0]] + b_end[yb[T_ - 1]];

#pragma unroll
    for (int m = 16; m >= 1; m >>= 1) e += __shfl_xor(e, m, 32);
    if (lane == 0) energy[b] = e;
}

// ---------------------------------------------------------------------------
// Kernel 2: alpha recursion. One wave (32 threads) handles 16 batch rows.
//
// D/C layout (16x16 f32): VGPR v, lane l<16 -> (row v,   col l)
//                                  lane l>=16 -> (row v+8, col l-16)
// A layout (16x4 f32):    lane m<16: vgpr0=p[m][k0], vgpr1=p[m][k0+1]
//                         lane m>=16: vgpr0=p[m-16][k0+2], vgpr1=p[m-16][k0+3]
// B layout (4x16 f32):    lane n<16: vgpr0=B[k0][n],  vgpr1=B[k0+1][n]
//                         lane n>=16: vgpr0=B[k0+2][n-16], vgpr1=B[k0+3][n-16]
// ---------------------------------------------------------------------------
__global__ __launch_bounds__(32) void crf_scan_kernel(
    const float* __restrict__ x, const float* __restrict__ U,
    const float* __restrict__ b_start, const float* __restrict__ b_end,
    const float* __restrict__ energy, float* __restrict__ out)
{
    __shared__ float pbuf[16 * C_];   // p staged row-major [16 rows][32 cols]

    const int lane = threadIdx.x & 31;
    const int half = lane >> 4;       // 0 or 1 (lane half)
    const int lcol = lane & 15;       // 0..15
    const int b0   = blockIdx.x * 16; // first batch row of this wave's block

#if HAVE_WMMA_F32X4
    // exp(U) pre-loaded into registers in B-matrix layout:
    // bU[tile][kk] covers K rows 4kk..4kk+3, N cols tile*16 + 0..15
    v2f bU[2][8];
#pragma unroll
    for (int tile = 0; tile < 2; ++tile) {
#pragma unroll
        for (int kk = 0; kk < 8; ++kk) {
            const int kr  = 4 * kk + 2 * half;
            const int col = tile * 16 + lcol;
            bU[tile][kk][0] = __expf(U[(kr + 0) * C_ + col]);
            bU[tile][kk][1] = __expf(U[(kr + 1) * C_ + col]);
        }
    }
#else
    // Scalar fallback: exp(U) columns (lcol and lcol+16) in registers.
    float eU0[32], eU1[32];
#pragma unroll
    for (int i = 0; i < 32; ++i) {
        eU0[i] = __expf(U[i * C_ + lcol]);
        eU1[i] = __expf(U[i * C_ + 16 + lcol]);
    }
#endif

    // alpha_0 = x[:,0,:] + b_start, held in D layout (a0: cols 0-15, a1: 16-31)
    float a0[8], a1[8];
#pragma unroll
    for (int v = 0; v < 8; ++v) {
        const size_t base = (size_t)(b0 + v + 8 * half) * TC_;
        a0[v] = x[base + lcol]      + b_start[lcol];
        a1[v] = x[base + 16 + lcol] + b_start[16 + lcol];
    }

    // Software-pipelined x_t loads (one iteration ahead).
    float xv0[8], xv1[8];
#pragma unroll
    for (int v = 0; v < 8; ++v) {
        const size_t base = (size_t)(b0 + v + 8 * half) * TC_ + (size_t)1 * C_;
        xv0[v] = x[base + lcol];
        xv1[v] = x[base + 16 + lcol];
    }

    for (int t = 1; t < T_; ++t) {
        // Issue next step's x loads now; consumed a full iteration later.
        const int tn = (t + 1 < T_) ? (t + 1) : (T_ - 1);
        float xn0[8], xn1[8];
#pragma unroll
        for (int v = 0; v < 8; ++v) {
            const size_t base = (size_t)(b0 + v + 8 * half) * TC_ + (size_t)tn * C_;
            xn0[v] = x[base + lcol];
            xn1[v] = x[base + 16 + lcol];
        }
        if (t + 8 < T_) {
            __builtin_prefetch(
                &x[(size_t)(b0 + lcol) * TC_ + (size_t)(t + 8) * C_ + half * 16], 0, 3);
        }

        // Row max over the 32 classes (masks <= 8 keep the shuffle inside each
        // 16-lane half, matching the D layout's row<->lane-half mapping).
        float m[8];
#pragma unroll
        for (int v = 0; v < 8; ++v) {
            float mv = fmaxf(a0[v], a1[v]);
            mv = fmaxf(mv, __shfl_xor(mv, 1, 32));
            mv = fmaxf(mv, __shfl_xor(mv, 2, 32));
            mv = fmaxf(mv, __shfl_xor(mv, 4, 32));
            mv = fmaxf(mv, __shfl_xor(mv, 8, 32));
            m[v] = mv;
        }

        // p = exp(alpha - m) staged to LDS row-major. Single wave: LDS ops
        // from one wave are processed in order, no barrier needed.
#pragma unroll
        for (int v = 0; v < 8; ++v) {
            const int r = v + 8 * half;
            pbuf[r * C_ + lcol]      = __expf(a0[v] - m[v]);
            pbuf[r * C_ + 16 + lcol] = __expf(a1[v] - m[v]);
        }

        // w[16x32] = p[16x32] * exp(U)[32x32], fp32 WMMA, K in chunks of 4.
        v8f acc0 = {};
        v8f acc1 = {};
#if HAVE_WMMA_F32X4
#pragma unroll
        for (int kk = 0; kk < 8; ++kk) {
            const v2f a2 = *(const v2f*)&pbuf[lcol * C_ + 4 * kk + 2 * half];
            acc0 = __builtin_amdgcn_wmma_f32_16x16x4_f32(
                false, a2, false, bU[0][kk], (short)0, acc0, false, false);
            acc1 = __builtin_amdgcn_wmma_f32_16x16x4_f32(
                false, a2, false, bU[1][kk], (short)0, acc1, false, false);
        }
#else
#pragma unroll
        for (int v = 0; v < 8; ++v) {
            const int r = v + 8 * half;
            float w0 = 0.f, w1 = 0.f;
#pragma unroll
            for (int i = 0; i < 32; ++i) {
                const float pi = pbuf[r * C_ + i];
                w0 = fmaf(pi, eU0[i], w0);
                w1 = fmaf(pi, eU1[i], w1);
            }
            acc0[v] = w0;
            acc1[v] = w1;
        }
#endif

        // alpha_new = m + ln(w) + x_t   (raw v_log_f32, ln2 folded into FMA)
#pragma unroll
        for (int v = 0; v < 8; ++v) {
#if defined(__AMDGCN__) && __has_builtin(__builtin_amdgcn_logf)
            a0[v] = fmaf(LN2F, __builtin_amdgcn_logf(acc0[v]), m[v] + xv0[v]);
            a1[v] = fmaf(LN2F, __builtin_amdgcn_logf(acc1[v]), m[v] + xv1[v]);
#else
            a0[v] = m[v] + __logf(acc0[v]) + xv0[v];
            a1[v] = m[v] + __logf(acc1[v]) + xv1[v];
#endif
            xv0[v] = xn0[v];
            xv1[v] = xn1[v];
        }
    }

    // b_end enters linearly after the log: fold it in once, after the loop.
    const float bend0 = b_end[lcol];
    const float bend1 = b_end[16 + lcol];
#pragma unroll
    for (int v = 0; v < 8; ++v) {
        a0[v] += bend0;
        a1[v] += bend1;
    }

    // log_z[b] = logsumexp_j(alpha_final[b,j]); out[b] = log_z - energy[b]
#pragma unroll
    for (int v = 0; v < 8; ++v) {
        float mv = fmaxf(a0[v], a1[v]);
        mv = fmaxf(mv, __shfl_xor(mv, 1, 32));
        mv = fmaxf(mv, __shfl_xor(mv, 2, 32));
        mv = fmaxf(mv, __shfl_xor(mv, 4, 32));
        mv = fmaxf(mv, __shfl_xor(mv, 8, 32));
        float s = __expf(a0[v] - mv) + __expf(a1[v] - mv);
        s += __shfl_xor(s, 1, 32);
        s += __shfl_xor(s, 2, 32);
        s += __shfl_xor(s, 4, 32);
        s += __shfl_xor(s, 8, 32);
        if (lcol == 0) {
            const int b = b0 + v + 8 * half;
            out[b] = (mv + fast_ln(s)) - energy[b];
        }
    }
}

// ---------------------------------------------------------------------------
extern "C" void kernel_launch(void* const* d_in, const int* in_sizes, int n_in,
                              void* d_out, int out_size, void* d_ws, size_t ws_size,
                              hipStream_t stream) {
    const float* x       = (const float*)d_in[0];
    const float* U       = (const float*)d_in[1];
    const float* b_start = (const float*)d_in[2];
    const float* b_end   = (const float*)d_in[3];
    const int*   y       = (const int*)d_in[4];
    float* out    = (float*)d_out;
    float* energy = (float*)d_ws;   // 128 floats of scratch

    crf_energy_kernel<<<B_ / 8, 256, 0, stream>>>(x, U, b_start, b_end, y, energy);
    crf_scan_kernel<<<B_ / 16, 32, 0, stream>>>(x, U, b_start, b_end, energy, out);
}